// SelfAttention_13391708029779
// MI455X (gfx1250) — compile-verified
//
#include <hip/hip_runtime.h>
#include <math.h>

// ---------------------------------------------------------------------------
// SelfAttention (attention over BATCH axis) for MI455X / gfx1250
//   x:[16,64,64,256] fp32.  f=x@Wf+bf [N,32], g=x@Wg+bg [N,32], h=x@Wh+bh [N,256]
//   s = G·F^T [16,16], beta = softmax(s), out = gamma*(beta@H) + x
// fp32 WMMA (V_WMMA_F32_16X16X4_F32) everywhere a matmul appears: the workload
// is HBM-bound (~350MB / 23.3 TB/s ≈ 15us) so fp32 matrix ops are free and
// match reference numerics. Weights are pre-packed into WMMA fragment order
// (coalesced b64 loads, 1 load/WMMA) and each wave register-blocks 4 row tiles
// so each B fragment feeds 4 WMMAs (weight L2 traffic cut 4x). x tiles are
// staged to LDS with gfx1250 async global->LDS b128 copies (ASYNCcnt).
// ---------------------------------------------------------------------------

typedef float v2f __attribute__((ext_vector_type(2)));
typedef float v4f __attribute__((ext_vector_type(4)));
typedef float v8f __attribute__((ext_vector_type(8)));
typedef int   v4i_ls __attribute__((vector_size(4 * sizeof(int))));   // matches builtin proto
typedef __attribute__((address_space(1))) v4i_ls* gptr_b128;
typedef __attribute__((address_space(3))) v4i_ls* lptr_b128;

#define HW_       4096      // 64*64
#define C_        256
#define C8_       32
#define B_        16
#define NPIX_     65536     // B_*HW_
#define HWC_      1048576   // HW_*C_
#define KFLAT_    131072    // HW_*C8_
#define SCHUNKS_  1024
#define SCHUNK_K_ 128       // KFLAT_/SCHUNKS_
#define NTILES_   20        // 16 (Wh) + 2 (Wf) + 2 (Wg) 16-wide column tiles
#define KSTEPS_   64        // C_/4

// ---------------------------------------------------------------------------
// Kernel 0: pack Wh/Wf/Wg into WMMA-B fragment order.
// pack[((tile*64 + kb)*32 + lane)] = float2{ W[(4kb+2*khalf)*ldw + n0+lh],
//                                            W[(4kb+2*khalf+1)*ldw + n0+lh] }
// -> in the GEMM, lane L's B fragment for k-step kb is one coalesced b64 load.
// ---------------------------------------------------------------------------
__global__ __launch_bounds__(256) void wpack_kernel(
    const float* __restrict__ Wf, const float* __restrict__ Wg,
    const float* __restrict__ Wh, float* __restrict__ wpack)
{
    const int idx = blockIdx.x * 256 + threadIdx.x;       // pair index
    if (idx >= NTILES_ * KSTEPS_ * 32) return;
    const int t    = idx / (KSTEPS_ * 32);
    const int r    = idx % (KSTEPS_ * 32);
    const int kb   = r / 32;
    const int lane = r % 32;
    const int lh    = lane & 15;
    const int khalf = lane >> 4;

    const float* W; int ldw, n0;
    if (t < 16)      { W = Wh; ldw = C_;  n0 = t * 16; }
    else if (t < 18) { W = Wf; ldw = C8_; n0 = (t - 16) * 16; }
    else             { W = Wg; ldw = C8_; n0 = (t - 18) * 16; }

    const int k0 = kb * 4 + khalf * 2;
    wpack[(size_t)idx * 2 + 0] = W[(size_t)(k0    ) * ldw + n0 + lh];
    wpack[(size_t)idx * 2 + 1] = W[(size_t)(k0 + 1) * ldw + n0 + lh];
}

// ---------------------------------------------------------------------------
// Kernel 1: fused f/g/h GEMMs, register-blocked.
// 1024 blocks x 640 threads (20 waves). Block stages a 64x256 x-tile (64KB)
// in LDS via async global->LDS b128. Wave w owns column tile w and computes
// FOUR 16x16 row tiles: 1 coalesced b64 weight load feeds 4 WMMAs per k-step.
// ---------------------------------------------------------------------------
__global__ __launch_bounds__(640) void fgh_gemm_kernel(
    const float* __restrict__ x, const float* __restrict__ wpack,
    const float* __restrict__ bf, const float* __restrict__ bg,
    const float* __restrict__ bh,
    float* __restrict__ h_ws, float* __restrict__ f_ws, float* __restrict__ g_ws)
{
    __shared__ float lds_x[64 * C_];                       // 64 KB of 320 KB WGP LDS
    const int tid = threadIdx.x;
    const unsigned row0 = blockIdx.x * 64u;

    const float* xt = x + (size_t)row0 * C_;               // 16384 contiguous floats
#if defined(__has_builtin) && __has_builtin(__builtin_amdgcn_global_load_async_to_lds_b128)
    for (int i = tid; i < (64 * C_) / 4; i += 640) {
        __builtin_amdgcn_global_load_async_to_lds_b128(
            (gptr_b128)(xt + (size_t)i * 4),
            (lptr_b128)&lds_x[i * 4],
            0, 0);
    }
    asm volatile("s_wait_asynccnt 0x0" ::: "memory");
#else
    for (int i = tid; i < (64 * C_) / 4; i += 640)
        *(v4f*)&lds_x[i * 4] = *(const v4f*)&xt[(size_t)i * 4];
#endif
    __syncthreads();

    const int wave  = tid >> 5;
    const int lane  = tid & 31;
    const int lh    = lane & 15;      // row (A) / col (B,D) within tile
    const int khalf = lane >> 4;      // 0: K={0,1}, 1: K={2,3}

    const float* bsrc; float* outp; int ldo, n0;
    if (wave < 16)      { bsrc = bh; n0 = wave * 16;        ldo = C_;  outp = h_ws + (size_t)row0 * C_  + n0; }
    else if (wave < 18) { bsrc = bf; n0 = (wave - 16) * 16; ldo = C8_; outp = f_ws + (size_t)row0 * C8_ + n0; }
    else                { bsrc = bg; n0 = (wave - 18) * 16; ldo = C8_; outp = g_ws + (size_t)row0 * C8_ + n0; }

    const v2f* wp = (const v2f*)wpack + (size_t)wave * (KSTEPS_ * 32) + lane;

    v8f acc0 = {}, acc1 = {}, acc2 = {}, acc3 = {};
    #pragma unroll 2
    for (int kb = 0; kb < KSTEPS_; ++kb) {
        const v2f b  = wp[kb * 32];                        // coalesced 256B/wave
        const int k0 = kb * 4 + khalf * 2;
        const v2f a0 = *(const v2f*)&lds_x[(      lh) * C_ + k0];
        const v2f a1 = *(const v2f*)&lds_x[(16  + lh) * C_ + k0];
        const v2f a2 = *(const v2f*)&lds_x[(32  + lh) * C_ + k0];
        const v2f a3 = *(const v2f*)&lds_x[(48  + lh) * C_ + k0];
        acc0 = __builtin_amdgcn_wmma_f32_16x16x4_f32(false, a0, false, b, (short)0, acc0, false, false);
        acc1 = __builtin_amdgcn_wmma_f32_16x16x4_f32(false, a1, false, b, (short)0, acc1, false, false);
        acc2 = __builtin_amdgcn_wmma_f32_16x16x4_f32(false, a2, false, b, (short)0, acc2, false, false);
        acc3 = __builtin_amdgcn_wmma_f32_16x16x4_f32(false, a3, false, b, (short)0, acc3, false, false);
    }

    const float bv = bsrc[n0 + lh];
    #pragma unroll
    for (int r = 0; r < 8; ++r) {
        const int m = r + khalf * 8;                       // D: vgpr r -> rows r, r+8
        outp[(size_t)(m     ) * ldo + lh] = acc0[r] + bv;
        outp[(size_t)(m + 16) * ldo + lh] = acc1[r] + bv;
        outp[(size_t)(m + 32) * ldo + lh] = acc2[r] + bv;
        outp[(size_t)(m + 48) * ldo + lh] = acc3[r] + bv;
    }
}

// ---------------------------------------------------------------------------
// Kernel 2: s-partials.  s = G[16,131072] @ F^T[131072,16]; K split into 1024
// fixed chunks of 128; one wave per block, 32 WMMAs, partial -> fixed slot
// (deterministic, no float atomics).
// ---------------------------------------------------------------------------
__global__ __launch_bounds__(32) void s_partial_kernel(
    const float* __restrict__ g_ws, const float* __restrict__ f_ws,
    float* __restrict__ part)
{
    const int    lane  = threadIdx.x & 31;
    const int    lh    = lane & 15;
    const int    khalf = lane >> 4;
    const size_t kbase = (size_t)blockIdx.x * SCHUNK_K_;

    const float* grow = g_ws + (size_t)lh * KFLAT_;        // A row  m = lh
    const float* frow = f_ws + (size_t)lh * KFLAT_;        // B col  n = lh (B = F^T)

    v8f acc = {};
    #pragma unroll 4
    for (int kb = 0; kb < SCHUNK_K_ / 4; ++kb) {
        const size_t k0 = kbase + (size_t)kb * 4 + khalf * 2;
        const v2f a = *(const v2f*)&grow[k0];
        const v2f b = *(const v2f*)&frow[k0];
        acc = __builtin_amdgcn_wmma_f32_16x16x4_f32(false, a, false, b, (short)0, acc, false, false);
    }

    float* p = part + (size_t)blockIdx.x * 256;
    #pragma unroll
    for (int r = 0; r < 8; ++r)
        p[(r + khalf * 8) * 16 + lh] = acc[r];
}

// ---------------------------------------------------------------------------
// Kernel 3: fixed-order reduction of 1024 partials + row softmax (one block).
// ---------------------------------------------------------------------------
__global__ __launch_bounds__(256) void softmax_kernel(
    const float* __restrict__ part, float* __restrict__ beta)
{
    const int t = threadIdx.x;
    float s = 0.0f;
    for (int i = 0; i < SCHUNKS_; ++i) s += part[i * 256 + t];

    __shared__ float sm[256];
    __shared__ float se[256];
    sm[t] = s;
    __syncthreads();

    const int row = t >> 4;
    float mx = -3.402823466e+38f;
    #pragma unroll
    for (int j = 0; j < 16; ++j) mx = fmaxf(mx, sm[row * 16 + j]);
    const float e = expf(s - mx);
    se[t] = e;
    __syncthreads();

    float d = 0.0f;
    #pragma unroll
    for (int j = 0; j < 16; ++j) d += se[row * 16 + j];
    beta[t] = e / d;
}

// ---------------------------------------------------------------------------
// Kernel 4: out = gamma * (beta @ H) + x.  Bandwidth kernel, float4; h is L2
// resident (ws ~85MB < 192MB L2). 16 h-vectors reused across all 16 outputs.
// ---------------------------------------------------------------------------
__global__ __launch_bounds__(256) void attn_out_kernel(
    const float* __restrict__ h_ws, const float* __restrict__ x,
    const float* __restrict__ beta, const float* __restrict__ gamma,
    float* __restrict__ out)
{
    __shared__ float sb[256];
    __shared__ float sg;
    const int t = threadIdx.x;
    sb[t] = beta[t];
    if (t == 0) sg = gamma[0];
    __syncthreads();

    const size_t col = ((size_t)blockIdx.x * 256 + t) * 4;

    v4f hv[16];
    #pragma unroll
    for (int bp = 0; bp < 16; ++bp)
        hv[bp] = *(const v4f*)&h_ws[(size_t)bp * HWC_ + col];

    const float gma = sg;
    #pragma unroll
    for (int b = 0; b < 16; ++b) {
        v4f acc = {};
        #pragma unroll
        for (int bp = 0; bp < 16; ++bp)
            acc += sb[b * 16 + bp] * hv[bp];
        const v4f xv = *(const v4f*)&x[(size_t)b * HWC_ + col];
        *(v4f*)&out[(size_t)b * HWC_ + col] = gma * acc + xv;
    }
}

// ---------------------------------------------------------------------------
// Workspace layout (floats):
//   h_ws  : 16,777,216 (64MB) | f_ws : 2,097,152 (8MB) | g_ws : 2,097,152 (8MB)
//   part  :    262,144 ( 1MB) | beta :        256      | wpack:   81,920 (320KB)
// total ~85.3 MB
// ---------------------------------------------------------------------------
extern "C" void kernel_launch(void* const* d_in, const int* in_sizes, int n_in,
                              void* d_out, int out_size, void* d_ws, size_t ws_size,
                              hipStream_t stream)
{
    const float* x     = (const float*)d_in[0];
    const float* Wf    = (const float*)d_in[1];
    const float* Wg    = (const float*)d_in[2];
    const float* Wh    = (const float*)d_in[3];
    const float* bf    = (const float*)d_in[4];
    const float* bg    = (const float*)d_in[5];
    const float* bh    = (const float*)d_in[6];
    const float* gamma = (const float*)d_in[7];
    float*       out   = (float*)d_out;

    float* wsf   = (float*)d_ws;
    float* h_ws  = wsf;
    float* f_ws  = h_ws + (size_t)NPIX_ * C_;
    float* g_ws  = f_ws + (size_t)NPIX_ * C8_;
    float* part  = g_ws + (size_t)NPIX_ * C8_;
    float* beta  = part + (size_t)SCHUNKS_ * 256;
    float* wpack = beta + 256;

    const int npairs = NTILES_ * KSTEPS_ * 32;             // 40960
    wpack_kernel<<<(npairs + 255) / 256, 256, 0, stream>>>(Wf, Wg, Wh, wpack);
    fgh_gemm_kernel<<<NPIX_ / 64, 640, 0, stream>>>(x, wpack, bf, bg, bh,
                                                    h_ws, f_ws, g_ws);
    s_partial_kernel<<<SCHUNKS_, 32, 0, stream>>>(g_ws, f_ws, part);
    softmax_kernel<<<1, 256, 0, stream>>>(part, beta);
    attn_out_kernel<<<HWC_ / 1024, 256, 0, stream>>>(h_ws, x, beta, gamma, out);

    (void)in_sizes; (void)n_in; (void)out_size; (void)ws_size;
}